// EdgeFeatureRGCN_56066503082346
// MI455X (gfx1250) — compile-verified
//
#include <hip/hip_runtime.h>
#include <math.h>

#define R_REL 8
#define NBASES 4
#define BN_EPS 1e-5f

typedef __attribute__((ext_vector_type(2))) float v2f;
typedef __attribute__((ext_vector_type(8))) float v8f;

// ---------------------------------------------------------------------------
// W build: Wt[(R+1)][out][in];  Wt[r][o][i] = sum_b comp[r,b]*bases[b][i][o];
//          Wt[R][o][i] = root[i][o]   (transposed for contiguous WMMA-B loads)
// ---------------------------------------------------------------------------
__global__ void build_w(const float* __restrict__ bases, const float* __restrict__ comp,
                        const float* __restrict__ root, float* __restrict__ Wt,
                        int in_dim, int out_dim) {
    int idx = blockIdx.x * blockDim.x + threadIdx.x;
    int per = out_dim * in_dim;
    int total = (R_REL + 1) * per;
    if (idx >= total) return;
    int r   = idx / per;
    int rem = idx - r * per;
    int o   = rem / in_dim;
    int i   = rem - o * in_dim;
    if (r < R_REL) {
        float a = 0.f;
#pragma unroll
        for (int b = 0; b < NBASES; ++b)
            a += comp[r * NBASES + b] * bases[((size_t)b * in_dim + i) * out_dim + o];
        Wt[idx] = a;
    } else {
        Wt[idx] = root[(size_t)i * out_dim + o];
    }
}

// ---------------------------------------------------------------------------
// Edge counts per (dst, relation) -- graph-invariant, computed once
// ---------------------------------------------------------------------------
__global__ void count_edges(const int* __restrict__ dst, const int* __restrict__ et,
                            float* __restrict__ cnt, int nedges) {
    int e = blockIdx.x * blockDim.x + threadIdx.x;
    if (e >= nedges) return;
    atomicAdd(&cnt[dst[e] * R_REL + et[e]], 1.0f);
}

// ---------------------------------------------------------------------------
// Scatter: agg[r][dst][:] += h[src][:] / max(cnt,1)   (float4 chunks per thread)
// ---------------------------------------------------------------------------
template <int IN_DIM>
__global__ void rgcn_scatter(const float* __restrict__ h,
                             const int* __restrict__ src, const int* __restrict__ dst,
                             const int* __restrict__ et, const float* __restrict__ cnt,
                             float* __restrict__ agg, int nedges, int nnodes) {
    constexpr int CH = IN_DIM / 4;
    int idx = blockIdx.x * blockDim.x + threadIdx.x;
    int total = nedges * CH;
    if (idx >= total) return;
    int e = idx / CH;
    int c = (idx - e * CH) * 4;
    int s = src[e], d = dst[e], r = et[e];
    float sc = 1.0f / fmaxf(cnt[d * R_REL + r], 1.0f);
    const float4 v = *reinterpret_cast<const float4*>(h + (size_t)s * IN_DIM + c);
    float* p = agg + ((size_t)r * nnodes + d) * IN_DIM + c;
    atomicAdd(p + 0, v.x * sc);
    atomicAdd(p + 1, v.y * sc);
    atomicAdd(p + 2, v.z * sc);
    atomicAdd(p + 3, v.w * sc);
}

// ---------------------------------------------------------------------------
// WMMA GEMM: out[n,:] = sum_r agg[r,n,:]*W_r + h[n,:]*root + bias
// Block = 64 rows x OUT_DIM cols; each wave owns a 16-col panel and computes
// 4 stacked 16x16 tiles, so each B float2 load feeds 4 WMMAs (4x B reuse,
// 4x fewer barriers vs. a 16-row tile). V_WMMA_F32_16X16X4_F32, exact f32.
// ---------------------------------------------------------------------------
template <int IN_DIM, int OUT_DIM>
__launch_bounds__(2 * OUT_DIM)
__global__ void rgcn_gemm(const float* __restrict__ agg, const float* __restrict__ h,
                          const float* __restrict__ Wt, const float* __restrict__ bias,
                          float* __restrict__ out, int nnodes) {
    constexpr int NWAVES = OUT_DIM / 16;
    constexpr int MT = 4;                 // stacked 16-row subtiles per block
    constexpr int LDA = IN_DIM + 4;       // pad: A reads are row-strided in LDS
    __shared__ float As[MT * 16 * LDA];

    const int tid  = threadIdx.x;
    const int wave = tid >> 5;
    const int lane = tid & 31;
    const int half = lane >> 4;  // K-half select for A/B operands
    const int lid  = lane & 15;  // M (A) / N (B,C,D) index
    const int m0   = blockIdx.x * (MT * 16);

    v8f acc[MT];
#pragma unroll
    for (int t = 0; t < MT; ++t)
        acc[t] = (v8f){0.f, 0.f, 0.f, 0.f, 0.f, 0.f, 0.f, 0.f};

    for (int rr = 0; rr <= R_REL; ++rr) {
        // cooperative stage of the 64 x IN_DIM A tile (shared by all waves)
        for (int idx = tid; idx < MT * 16 * IN_DIM; idx += 32 * NWAVES) {
            int row = idx / IN_DIM, col = idx - row * IN_DIM;
            int g = m0 + row;
            if (g >= nnodes) g = nnodes - 1;  // clamp; extra rows masked at store
            float v = (rr < R_REL) ? agg[((size_t)rr * nnodes + g) * IN_DIM + col]
                                   : h[(size_t)g * IN_DIM + col];
            As[row * LDA + col] = v;
        }
        __syncthreads();

        const float* Bt = Wt + ((size_t)rr * OUT_DIM + wave * 16 + lid) * IN_DIM;
#pragma unroll
        for (int k0 = 0; k0 < IN_DIM; k0 += 4) {
            int k = k0 + 2 * half;
            v2f b; b.x = Bt[k]; b.y = Bt[k + 1];  // B: 4x16, lane = N, K split by half
#pragma unroll
            for (int t = 0; t < MT; ++t) {
                const float* Arow = As + (t * 16 + lid) * LDA;
                v2f a; a.x = Arow[k]; a.y = Arow[k + 1];  // A: 16x4 per subtile
                acc[t] = __builtin_amdgcn_wmma_f32_16x16x4_f32(
                    false, a, false, b, (short)0, acc[t], false, false);
            }
        }
        __syncthreads();
    }

    const int ncol = wave * 16 + lid;
    const float bv = bias[ncol];
#pragma unroll
    for (int t = 0; t < MT; ++t) {
#pragma unroll
        for (int j = 0; j < 8; ++j) {  // D: VGPR j -> row j (half 0) / j+8 (half 1)
            int mrow = m0 + t * 16 + j + 8 * half;
            if (mrow < nnodes) out[(size_t)mrow * OUT_DIM + ncol] = acc[t][j] + bv;
        }
    }
}

// ---------------------------------------------------------------------------
// BatchNorm stats (sum / sumsq per channel) + fused apply + leaky ReLU
// ---------------------------------------------------------------------------
__global__ void bn_stats(const float* __restrict__ h, float* __restrict__ sums,
                         int n, int C) {
    int c = threadIdx.x;  // blockDim.x == C (coalesced across channels)
    float s = 0.f, ss = 0.f;
    for (int row = blockIdx.x; row < n; row += gridDim.x) {
        float v = h[(size_t)row * C + c];
        s += v; ss += v * v;
    }
    atomicAdd(&sums[c], s);
    atomicAdd(&sums[C + c], ss);
}

__global__ void bn_apply_leaky(float* __restrict__ h, const float* __restrict__ sums,
                               const float* __restrict__ g, const float* __restrict__ b,
                               int n, int C) {
    int idx = blockIdx.x * blockDim.x + threadIdx.x;
    if (idx >= n * C) return;
    int c = idx % C;
    float invN = 1.0f / (float)n;
    float mean = sums[c] * invN;
    float var  = sums[C + c] * invN - mean * mean;
    float v = (h[idx] - mean) * rsqrtf(var + BN_EPS) * g[c] + b[c];
    h[idx] = (v >= 0.f) ? v : 0.1f * v;
}

// ---------------------------------------------------------------------------
// Row-wise L2 normalize (C = 64), in place
// ---------------------------------------------------------------------------
__global__ void l2norm_inplace(float* __restrict__ h) {
    __shared__ float red[64];
    int row = blockIdx.x, t = threadIdx.x;
    float v = h[(size_t)row * 64 + t];
    red[t] = v * v;
    __syncthreads();
    for (int s = 32; s > 0; s >>= 1) {
        if (t < s) red[t] += red[t + s];
        __syncthreads();
    }
    float nrm = fmaxf(sqrtf(red[0]), 1e-12f);
    h[(size_t)row * 64 + t] = v / nrm;
}

// ---------------------------------------------------------------------------
extern "C" void kernel_launch(void* const* d_in, const int* in_sizes, int n_in,
                              void* d_out, int out_size, void* d_ws, size_t ws_size,
                              hipStream_t stream) {
    (void)n_in; (void)out_size; (void)ws_size;
    const float* x      = (const float*)d_in[0];
    const int*   eidx   = (const int*)d_in[1];
    const int*   etype  = (const int*)d_in[2];
    const float* bases1 = (const float*)d_in[3];
    const float* comp1  = (const float*)d_in[4];
    const float* root1  = (const float*)d_in[5];
    const float* bias1  = (const float*)d_in[6];
    const float* g1     = (const float*)d_in[7];
    const float* b1     = (const float*)d_in[8];
    const float* bases2 = (const float*)d_in[9];
    const float* comp2  = (const float*)d_in[10];
    const float* root2  = (const float*)d_in[11];
    const float* bias2  = (const float*)d_in[12];
    const float* g2     = (const float*)d_in[13];
    const float* b2     = (const float*)d_in[14];
    const float* bases3 = (const float*)d_in[15];
    const float* comp3  = (const float*)d_in[16];
    const float* root3  = (const float*)d_in[17];
    const float* bias3  = (const float*)d_in[18];

    const int IN = 64, H = 128, OUT = 64;
    const int N = in_sizes[0] / IN;
    const int E = in_sizes[2];
    const int* srcv = eidx;
    const int* dstv = eidx + E;

    // workspace carve-out (256B aligned)
    char* ws = (char*)d_ws;
    size_t off = 0;
    auto carve = [&](size_t bytes) {
        void* p = ws + off;
        off = (off + bytes + 255) & ~(size_t)255;
        return p;
    };
    float* agg  = (float*)carve((size_t)R_REL * N * H * 4);  // largest layer (in=128)
    float* h1   = (float*)carve((size_t)N * H * 4);
    float* h2   = (float*)carve((size_t)N * H * 4);
    float* cnt  = (float*)carve((size_t)N * R_REL * 4);
    float* Wt   = (float*)carve((size_t)(R_REL + 1) * H * H * 4);
    float* sums = (float*)carve((size_t)2 * H * 4);
    float* outp = (float*)d_out;

    const int MROWS = 64;  // GEMM block M-tile

    // --- shared across layers: per-(dst,relation) edge counts ---
    hipMemsetAsync(cnt, 0, (size_t)N * R_REL * 4, stream);
    count_edges<<<(E + 255) / 256, 256, 0, stream>>>(dstv, etype, cnt, E);

    // ---------------- Layer 1: 64 -> 128 ----------------
    build_w<<<((R_REL + 1) * H * IN + 255) / 256, 256, 0, stream>>>(bases1, comp1, root1, Wt, IN, H);
    hipMemsetAsync(agg, 0, (size_t)R_REL * N * IN * 4, stream);
    rgcn_scatter<64><<<(E * (IN / 4) + 255) / 256, 256, 0, stream>>>(x, srcv, dstv, etype, cnt, agg, E, N);
    rgcn_gemm<64, 128><<<(N + MROWS - 1) / MROWS, 256, 0, stream>>>(agg, x, Wt, bias1, h1, N);
    hipMemsetAsync(sums, 0, 2 * H * 4, stream);
    bn_stats<<<512, H, 0, stream>>>(h1, sums, N, H);
    bn_apply_leaky<<<(N * H + 255) / 256, 256, 0, stream>>>(h1, sums, g1, b1, N, H);

    // ---------------- Layer 2: 128 -> 128 ----------------
    build_w<<<((R_REL + 1) * H * H + 255) / 256, 256, 0, stream>>>(bases2, comp2, root2, Wt, H, H);
    hipMemsetAsync(agg, 0, (size_t)R_REL * N * H * 4, stream);
    rgcn_scatter<128><<<(E * (H / 4) + 255) / 256, 256, 0, stream>>>(h1, srcv, dstv, etype, cnt, agg, E, N);
    rgcn_gemm<128, 128><<<(N + MROWS - 1) / MROWS, 256, 0, stream>>>(agg, h1, Wt, bias2, h2, N);
    hipMemsetAsync(sums, 0, 2 * H * 4, stream);
    bn_stats<<<512, H, 0, stream>>>(h2, sums, N, H);
    bn_apply_leaky<<<(N * H + 255) / 256, 256, 0, stream>>>(h2, sums, g2, b2, N, H);

    // ---------------- Layer 3: 128 -> 64 + L2 normalize ----------------
    build_w<<<((R_REL + 1) * OUT * H + 255) / 256, 256, 0, stream>>>(bases3, comp3, root3, Wt, H, OUT);
    hipMemsetAsync(agg, 0, (size_t)R_REL * N * H * 4, stream);
    rgcn_scatter<128><<<(E * (H / 4) + 255) / 256, 256, 0, stream>>>(h2, srcv, dstv, etype, cnt, agg, E, N);
    rgcn_gemm<128, 64><<<(N + MROWS - 1) / MROWS, 128, 0, stream>>>(agg, h2, Wt, bias3, outp, N);
    l2norm_inplace<<<N, 64, 0, stream>>>(outp);
}